// GCN_test_73512660238663
// MI455X (gfx1250) — compile-verified
//
#include <hip/hip_runtime.h>
#include <hip/hip_bf16.h>
#include <stdint.h>

// ---------------------------------------------------------------------------
// GCNConv (normalize=True) + bias + ReLU for MI455X (gfx1250, wave32, WMMA).
//   out[d] = relu( dinv[d] * ( y[d] + sum_{(s->d)} y[s] ) + b ),
//   y[i]   = dinv[i] * (X @ W)[i],   dinv[i] = rsqrt(1 + indeg[i])
// ---------------------------------------------------------------------------

typedef float v2f __attribute__((ext_vector_type(2)));
typedef float v8f __attribute__((ext_vector_type(8)));

constexpr int N_NODES = 100000;
constexpr int N_EDGES = 1600000;
constexpr int IN_C    = 128;
constexpr int OUT_C   = 64;
constexpr int M_TILES = N_NODES / 16;        // 6250 exactly
constexpr int WSTRIDE = 132;                 // LDS row stride (floats) for W^T:
                                             // 132 % 64 == 4 -> conflict-free b64
                                             // B-fragment reads; even -> 8B align

// ---------------- degree / normalization ----------------------------------

__global__ void gcn_deg_init(float* __restrict__ deg) {
    int i = blockIdx.x * blockDim.x + threadIdx.x;
    if (i < N_NODES) deg[i] = 1.0f;          // self-loop
}

__global__ void gcn_deg_count(const long long* __restrict__ edge_index,
                              float* __restrict__ deg) {
    int e = blockIdx.x * blockDim.x + threadIdx.x;
    if (e < N_EDGES) {
        int d = (int)edge_index[N_EDGES + e];   // row 1 = dst
        atomicAdd(&deg[d], 1.0f);
    }
}

__global__ void gcn_deg_rsqrt(float* __restrict__ deg) {
    int i = blockIdx.x * blockDim.x + threadIdx.x;
    if (i < N_NODES) deg[i] = __frsqrt_rn(deg[i]);   // deg >= 1 always
}

// ---------------- X @ W with f32 WMMA, fused dinv row scale ----------------
// One wave computes a 16(M) x 64(N) strip. K = 128 in chunks of 4 via
// V_WMMA_F32_16X16X4_F32. W is staged TRANSPOSED in LDS so each B fragment
// {W[ka][n], W[ka+1][n]} is one contiguous ds_load_b64 (no repack moves).

__global__ void __launch_bounds__(256)
gcn_gemm_wmma(const float* __restrict__ X,      // [N,128]
              const float* __restrict__ W,      // [128,64] row-major
              const float* __restrict__ dinv,   // [N]
              float* __restrict__ Y,            // [N,64] scratch (read-only later)
              float* __restrict__ Acc) {        // [N,64] accumulator (= d_out)
    __shared__ float sWt[OUT_C * WSTRIDE];      // W^T, padded rows: ~33 KB

    // cooperative transpose fill: sWt[n*WSTRIDE + k] = W[k*64 + n]
    for (int i = threadIdx.x; i < IN_C * OUT_C; i += blockDim.x) {
        const int k = i >> 6;                   // 0..127 (coalesced global read)
        const int n = i & 63;                   // 0..63
        sWt[n * WSTRIDE + k] = W[i];
    }
    __syncthreads();

    const int wave  = threadIdx.x >> 5;
    const int lane  = threadIdx.x & 31;
    const int tile  = blockIdx.x * (blockDim.x >> 5) + wave;
    if (tile >= M_TILES) return;                // uniform per-wave: EXEC stays all-1s

    const int mbase = tile * 16;
    const int m     = lane & 15;                // A row index
    const int n     = lane & 15;                // B/D column index
    const int half  = lane >> 4;                // 0: K{0,1}/M{0..7}, 1: K{2,3}/M{8..15}

    v8f acc0 = {}, acc1 = {}, acc2 = {}, acc3 = {};

    const float* __restrict__ arow = X + (size_t)(mbase + m) * IN_C;
    const float* __restrict__ bt0  = sWt + (n +  0) * WSTRIDE + half * 2;
    const float* __restrict__ bt1  = sWt + (n + 16) * WSTRIDE + half * 2;
    const float* __restrict__ bt2  = sWt + (n + 32) * WSTRIDE + half * 2;
    const float* __restrict__ bt3  = sWt + (n + 48) * WSTRIDE + half * 2;

#pragma unroll
    for (int kc = 0; kc < IN_C; kc += 4) {
        const int ka = kc + half * 2;
        // A fragment: lane holds X[mbase+m][ka], X[mbase+m][ka+1]
        v2f a  = *(const v2f*)(arow + ka);
        // B fragments: contiguous pairs in transposed LDS -> single b64 each
        v2f b0 = *(const v2f*)(bt0 + kc);
        v2f b1 = *(const v2f*)(bt1 + kc);
        v2f b2 = *(const v2f*)(bt2 + kc);
        v2f b3 = *(const v2f*)(bt3 + kc);

        acc0 = __builtin_amdgcn_wmma_f32_16x16x4_f32(false, a, false, b0,
                                                     (short)0, acc0, false, false);
        acc1 = __builtin_amdgcn_wmma_f32_16x16x4_f32(false, a, false, b1,
                                                     (short)0, acc1, false, false);
        acc2 = __builtin_amdgcn_wmma_f32_16x16x4_f32(false, a, false, b2,
                                                     (short)0, acc2, false, false);
        acc3 = __builtin_amdgcn_wmma_f32_16x16x4_f32(false, a, false, b3,
                                                     (short)0, acc3, false, false);
    }

    // Epilogue: D VGPR v holds row M = v + half*8, col = n (ISA 7.12.2).
    // y = dinv[row] * acc; seed the output accumulator with the self-loop term.
#pragma unroll
    for (int v = 0; v < 8; ++v) {
        const int row  = mbase + v + half * 8;
        const float s  = dinv[row];
        const size_t o = (size_t)row * OUT_C + n;
        const float y0 = s * acc0[v], y1 = s * acc1[v];
        const float y2 = s * acc2[v], y3 = s * acc3[v];
        Y[o +  0] = y0;  Acc[o +  0] = y0;
        Y[o + 16] = y1;  Acc[o + 16] = y1;
        Y[o + 32] = y2;  Acc[o + 32] = y2;
        Y[o + 48] = y3;  Acc[o + 48] = y3;
    }
}

// ---------------- edge scatter: Acc[dst] += y[src] -------------------------
// 32 threads per edge, float2 per thread (y is L2-resident: 25.6 MB << 192 MB).

__global__ void gcn_scatter(const long long* __restrict__ edge_index,
                            const float* __restrict__ Y,
                            float* __restrict__ Acc) {
    int t = blockIdx.x * blockDim.x + threadIdx.x;   // < N_EDGES*32 = 51.2M
    int e = t >> 5;
    int c = (t & 31) * 2;
    int s = (int)edge_index[e];
    int d = (int)edge_index[N_EDGES + e];
    float2 v = *(const float2*)(Y + (size_t)s * OUT_C + c);
    atomicAdd(Acc + (size_t)d * OUT_C + c,     v.x);
    atomicAdd(Acc + (size_t)d * OUT_C + c + 1, v.y);
}

// ---------------- finalize: out = relu(dinv * acc + b) ---------------------

__global__ void gcn_finalize(float* __restrict__ Out,
                             const float* __restrict__ dinv,
                             const float* __restrict__ b) {
    int idx = blockIdx.x * blockDim.x + threadIdx.x;
    if (idx < N_NODES * OUT_C) {
        int i = idx >> 6;       // node
        int c = idx & 63;       // channel
        float v = dinv[i] * Out[idx] + b[c];
        Out[idx] = fmaxf(v, 0.0f);
    }
}

// ---------------------------------------------------------------------------

extern "C" void kernel_launch(void* const* d_in, const int* in_sizes, int n_in,
                              void* d_out, int out_size, void* d_ws, size_t ws_size,
                              hipStream_t stream) {
    const float*     node_emb = (const float*)d_in[0];       // [100000,128]
    const long long* edge_idx = (const long long*)d_in[1];   // [2,1600000] int64
    const float*     W        = (const float*)d_in[2];       // [128,64]
    const float*     bias     = (const float*)d_in[3];       // [64]
    float*           out      = (float*)d_out;               // [100000,64]

    // workspace: dinv (N f32, padded) | Y (N*64 f32)
    float* dinv = (float*)d_ws;
    float* Y    = dinv + 100096;   // 100000 rounded up to multiple of 256

    // 1) degree (with self-loop) -> dinv = rsqrt(deg), in place
    gcn_deg_init <<<(N_NODES + 255) / 256, 256, 0, stream>>>(dinv);
    gcn_deg_count<<<(N_EDGES + 255) / 256, 256, 0, stream>>>(edge_idx, dinv);
    gcn_deg_rsqrt<<<(N_NODES + 255) / 256, 256, 0, stream>>>(dinv);

    // 2) y = dinv * (X @ W) via f32 WMMA; also seeds Acc(=out) with self-loop term
    {
        const int waves_per_block = 8;                        // 256 threads
        const int blocks = (M_TILES + waves_per_block - 1) / waves_per_block; // 782
        gcn_gemm_wmma<<<blocks, 256, 0, stream>>>(node_emb, W, dinv, Y, out);
    }

    // 3) Acc[dst] += y[src] over all edges (L2-resident gather + f32 atomics)
    {
        const int total = N_EDGES * 32;                       // 51.2M threads
        gcn_scatter<<<total / 256, 256, 0, stream>>>(edge_idx, Y, out);
    }

    // 4) out = relu(dinv * Acc + b), in place
    gcn_finalize<<<(N_NODES * OUT_C + 255) / 256, 256, 0, stream>>>(out, dinv, bias);
}